// Downsample1d_44298292691471
// MI455X (gfx1250) — compile-verified
//
#include <hip/hip_runtime.h>
#include <cstdint>
#include <cstddef>

// Downsample1d: depthwise FIR [1/8,3/8,3/8,1/8], stride 2, reflect pad 1.
// x: [8,128,65536] f32  ->  out: [8,128,32768] f32
//
// out[t] = w0*x[m(2t-1)] + w1*x[2t] + w2*x[2t+1] + w3*x[m(2t+2)]   (m = reflect)

#define TT    65536          // input length
#define TOUT  32768          // output length
#define ROWS  1024           // B*C rows
#define RG    16             // rows per group (WMMA M)
#define NG    (ROWS / RG)    // 64 row groups
#define NOUT  256            // outputs per row per block
#define NTILE (TOUT / NOUT)  // 128 tiles along T
#define THREADS 128          // 4 waves (wave32)
#define PIN   520            // staged floats per row: p in [-4, 516)
#define LROW  585            // LDS row pitch in floats after TDM padding (520/8*9)

typedef __attribute__((ext_vector_type(2))) float v2f;
typedef __attribute__((ext_vector_type(8))) float v8f;
typedef __attribute__((ext_vector_type(4))) unsigned int v4u;
typedef __attribute__((ext_vector_type(8))) int v8i;
typedef __attribute__((ext_vector_type(4))) int v4i;

// Existence checks only make sense in the DEVICE pass; __has_builtin for
// amdgcn builtins reports false in hipcc's host pass.
#if defined(__HIP_DEVICE_COMPILE__)
#if !__has_builtin(__builtin_amdgcn_tensor_load_to_lds)
#error "device: tensor_load_to_lds builtin not available"
#endif
#if !__has_builtin(__builtin_amdgcn_s_wait_tensorcnt)
#error "device: s_wait_tensorcnt builtin not available"
#endif
#if !__has_builtin(__builtin_amdgcn_wmma_f32_16x16x4_f32)
#error "device: wmma_f32_16x16x4_f32 builtin not available"
#endif
#endif

// logical staged column -> padded LDS column (1 pad DW after every 8 data DW)
__device__ __forceinline__ int lcol(int c) { return ((c >> 3) * 9) + (c & 7); }

__device__ __forceinline__ uint32_t lds_byte_addr(const void* p) {
#if defined(__HIP_DEVICE_COMPILE__)
  // generic (__shared__) pointer -> LDS byte offset
  return (uint32_t)(uintptr_t)(__attribute__((address_space(3))) const void*)p;
#else
  return (uint32_t)(uintptr_t)p;  // host parse only
#endif
}

__global__ __launch_bounds__(THREADS)
void downsample1d_fir_kernel(const float* __restrict__ x,
                             const float* __restrict__ kern,
                             float* __restrict__ out) {
  __shared__ __align__(16) float sh[RG * LROW];

  const int blk  = blockIdx.x;
  const int g    = blk / NTILE;      // row group
  const int tile = blk % NTILE;      // tile along T
  const int row0 = g * RG;
  const int t0   = tile * NOUT;
  const int gcs  = 2 * t0 - 4;       // global column of staged p = -4
  const int tid  = threadIdx.x;
  const int wave = tid >> 5;
  const int lane = tid & 31;
  const int h    = (lane >> 4) & 1;  // lane half (selects K pair / M+8)
  const int n    = lane & 15;        // N (output column within 16-tile)

  const float w0 = kern[0], w1 = kern[1], w2 = kern[2], w3 = kern[3];

  const bool interior = (tile != 0) && (tile != NTILE - 1);

  if (interior) {
    if (wave == 0) {
      // ---- Tensor DMA: 16 rows x 520 f32, row stride TT, global -> LDS ----
      const uint64_t ga = (uint64_t)(uintptr_t)(x + (size_t)row0 * TT + gcs);
      v4u g0;
      g0[0] = 1u;                                            // count=1 (user)
      g0[1] = lds_byte_addr(sh);                             // lds_addr
      g0[2] = (uint32_t)ga;                                  // global addr lo
      g0[3] = (uint32_t)((ga >> 32) & 0x01FFFFFFu) | (2u << 30); // hi | type=2
      v8i g1;
      g1[0] = (int)((2u << 16)      // data_size = 4B
                  | (1u << 20)      // pad_enable
                  | (2u << 22));    // pad_interval = 8 DW (pad_amount code 0 = 1 DW)
      g1[1] = (int)0xFFFF0000u;     // atomic_barrier=0 ; tensor_dim0 lo16 (huge)
      g1[2] = (int)0xFFFF7FFFu;     // tensor_dim0 hi16 ; tensor_dim1 lo16 (huge)
      g1[3] = (int)(((uint32_t)PIN << 16) | 0x7FFFu); // tile_dim0=520 ; tensor_dim1 hi16
      g1[4] = (int)RG;              // tile_dim1=16 ; tile_dim2=0
      g1[5] = (int)TT;              // tensor_dim0_stride lo32 = 65536
      g1[6] = 0;                    // stride hi ; tensor_dim1_stride lo = 0
      g1[7] = 0;
      v4i gz = {0, 0, 0, 0};
      (void)gz;
#if defined(__HIP_DEVICE_COMPILE__)
#if __clang_major__ >= 23
      v8i gz8 = {0, 0, 0, 0, 0, 0, 0, 0};
      __builtin_amdgcn_tensor_load_to_lds(g0, g1, gz, gz, gz8, 0);
#else
      __builtin_amdgcn_tensor_load_to_lds(g0, g1, gz, gz, 0);
#endif
      __builtin_amdgcn_s_wait_tensorcnt(0);
#endif
      (void)g0; (void)g1;
    }
  } else {
    // ---- boundary tiles: plain loader with reflect clamp ----
    for (int i = tid; i < RG * PIN; i += THREADS) {
      const int m   = i / PIN;
      const int col = i - m * PIN;
      int q = gcs + col;
      q = (q < 0) ? -q : q;                 // reflect at left:  m(-1)=1
      q = (q >= TT) ? (2 * TT - 2 - q) : q; // reflect at right: m(T)=T-2
      sh[m * LROW + lcol(col)] = x[(size_t)(row0 + m) * TT + q];
    }
  }
  __syncthreads();

  // ---- precompute banded B (depends only on lane & chunk c) ----
  // chunk c supplies input positions p = 4c-4+k (k = 2h, 2h+1 for this lane).
  // even p: w1 -> j=p/2, w3 -> j=p/2-1 ; odd p: w0 -> j=(p+1)/2, w2 -> j=(p-1)/2
  v2f Bv[10];
#pragma unroll
  for (int c = 0; c < 10; ++c) {
    float bx, by;
    if (h == 0) { // k=0 (p=4c-4 even), k=1 (p=4c-3 odd)
      bx = (n == 2 * c - 2) ? w1 : (n == 2 * c - 3) ? w3 : 0.0f;
      by = (n == 2 * c - 1) ? w0 : (n == 2 * c - 2) ? w2 : 0.0f;
    } else {      // k=2 (p=4c-2 even), k=3 (p=4c-1 odd)
      bx = (n == 2 * c - 1) ? w1 : (n == 2 * c - 2) ? w3 : 0.0f;
      by = (n == 2 * c)     ? w0 : (n == 2 * c - 1) ? w2 : 0.0f;
    }
    Bv[c].x = bx;
    Bv[c].y = by;
  }

  // A-gather addressing: staged col = 2*dt + 4c + 2h, with 2*dt % 8 == 0 and
  // 4c+2h never crossing a pad chunk, so padded addr is affine:
  //   m*LROW + (dt>>2)*9 + 2h + LC[c],  LC[c] = lcol(4c)  (compile-time)
  const int m = lane & 15;                 // A-matrix row held by this lane
  const float* abase = &sh[m * LROW + 2 * h];
  constexpr int LC[10] = {0, 4, 9, 13, 18, 22, 27, 31, 36, 40};

#pragma unroll
  for (int it = 0; it < 4; ++it) {
    const int dt = wave * 64 + it * 16;    // output offset of this 16-tile
    const float* p = abase + (wave * 144 + it * 36);  // (dt>>2)*9
    v8f acc = {0.f, 0.f, 0.f, 0.f, 0.f, 0.f, 0.f, 0.f};
#pragma unroll
    for (int c = 0; c < 10; ++c) {
      v2f a;
      a.x = p[LC[c]];
      a.y = p[LC[c] + 1];
#if defined(__HIP_DEVICE_COMPILE__)
      // D = A(16x4) * B(4x16) + C   (f32 WMMA, wave32)
      acc = __builtin_amdgcn_wmma_f32_16x16x4_f32(
          false, a, false, Bv[c], (short)0, acc, false, false);
#else
      acc[0] += a.x * Bv[c].x + a.y * Bv[c].y;  // host parse only
#endif
    }
    // D layout: VGPR r -> M=r (lanes 0-15), M=r+8 (lanes 16-31); N = lane%16
    const size_t obase = (size_t)row0 * TOUT + (size_t)(t0 + dt + n);
#pragma unroll
    for (int r = 0; r < 8; ++r) {
      out[obase + (size_t)(r + 8 * h) * TOUT] = acc[r];
    }
  }
}

extern "C" void kernel_launch(void* const* d_in, const int* in_sizes, int n_in,
                              void* d_out, int out_size, void* d_ws, size_t ws_size,
                              hipStream_t stream) {
  (void)in_sizes; (void)n_in; (void)d_ws; (void)ws_size; (void)out_size;
  const float* x    = (const float*)d_in[0];
  const float* kern = (const float*)d_in[1];
  float* out        = (float*)d_out;
  downsample1d_fir_kernel<<<dim3(NG * NTILE), dim3(THREADS), 0, stream>>>(x, kern, out);
}